// SpatialTransformer_23527830847641
// MI455X (gfx1250) — compile-verified
//
#include <hip/hip_runtime.h>

// SpatialTransformer: out = grid_sample_bilinear_border(img, base_grid + dvf)
// img: [B,C,H,W] f32, dvf: [B,2,H,W] f32, out: [B,C,H,W] f32
// B=16, C=4, H=W=512.
//
// Memory-bound gather: 1 thread per (b,y,x) output pixel computes all C=4
// channels, issuing 16 independent global_load_b32 gathers (4 taps x 4
// channels) so the memory system can overlap them. dvf reads and out writes
// are marked non-temporal so the streamed data does not evict the 64MB image
// working set from the 192MB L2 (img is re-read ~4x and should stay resident).

#ifndef STK_B
#define STK_B 16
#define STK_C 4
#define STK_H 512
#define STK_W 512
#endif

__global__ __launch_bounds__(256) void
spatial_transformer_bilinear_border(const float* __restrict__ img,
                                    const float* __restrict__ dvf,
                                    float* __restrict__ out) {
    constexpr int W  = STK_W;
    constexpr int H  = STK_H;
    constexpr int C  = STK_C;
    constexpr int HW = H * W;

    const int tid = blockIdx.x * blockDim.x + threadIdx.x;   // 0 .. B*H*W-1
    const int x = tid & (W - 1);
    const int y = (tid >> 9) & (H - 1);
    const int b = tid >> 18;

    // ---- displacement field (streamed once -> non-temporal loads) ----
    const int sp = y * W + x;                 // spatial offset within a plane
    const float* dvf_b = dvf + (size_t)b * 2 * HW + sp;
    const float dx = __builtin_nontemporal_load(dvf_b);        // channel 0
    const float dy = __builtin_nontemporal_load(dvf_b + HW);   // channel 1

    // ---- coordinates: match reference math ----
    // base grid: linspace(-1,1,N)[i] = -1 + i * (2/(N-1))
    const float gx = fmaf((float)x, 2.0f / (float)(W - 1), -1.0f) + dx;
    const float gy = fmaf((float)y, 2.0f / (float)(H - 1), -1.0f) + dy;
    // unnormalize (align_corners=True) then border clamp
    float px = (gx + 1.0f) * (0.5f * (float)(W - 1));
    float py = (gy + 1.0f) * (0.5f * (float)(H - 1));
    px = fminf(fmaxf(px, 0.0f), (float)(W - 1));
    py = fminf(fmaxf(py, 0.0f), (float)(H - 1));

    const float x0f = floorf(px);
    const float y0f = floorf(py);
    const int   x0  = (int)x0f;
    const int   y0  = (int)y0f;
    const int   x1  = min(x0 + 1, W - 1);
    const int   y1  = min(y0 + 1, H - 1);
    const float wx  = px - x0f;
    const float wy  = py - y0f;

    // ---- gather 4 taps per channel; 16 independent loads in flight ----
    const float* img_b = img + (size_t)b * C * HW;
    const int r0 = y0 * W;
    const int r1 = y1 * W;

    float v00[C], v01[C], v10[C], v11[C];
#pragma unroll
    for (int c = 0; c < C; ++c) {
        const float* p = img_b + c * HW;
        v00[c] = p[r0 + x0];
        v01[c] = p[r0 + x1];
        v10[c] = p[r1 + x0];
        v11[c] = p[r1 + x1];
    }

    // ---- bilinear blend + streaming (non-temporal) stores ----
    float* out_b = out + (size_t)b * C * HW + sp;
#pragma unroll
    for (int c = 0; c < C; ++c) {
        const float top = fmaf(wx, v01[c] - v00[c], v00[c]);
        const float bot = fmaf(wx, v11[c] - v10[c], v10[c]);
        const float res = fmaf(wy, bot - top, top);
        __builtin_nontemporal_store(res, out_b + c * HW);
    }
}

extern "C" void kernel_launch(void* const* d_in, const int* in_sizes, int n_in,
                              void* d_out, int out_size, void* d_ws, size_t ws_size,
                              hipStream_t stream) {
    (void)in_sizes; (void)n_in; (void)out_size; (void)d_ws; (void)ws_size;
    const float* img = (const float*)d_in[0];   // [16,4,512,512] f32
    const float* dvf = (const float*)d_in[1];   // [16,2,512,512] f32
    float* out = (float*)d_out;                 // [16,4,512,512] f32

    constexpr int total = STK_B * STK_H * STK_W;      // one thread per pixel
    constexpr int block = 256;                        // 8 wave32 per block
    const int grid = (total + block - 1) / block;     // 16384 blocks
    spatial_transformer_bilinear_border<<<grid, block, 0, stream>>>(img, dvf, out);
}